// GraphSAGEPlusPlusDA_65575560675419
// MI455X (gfx1250) — compile-verified
//
#include <hip/hip_runtime.h>
#include <hip/hip_bf16.h>
#include <cstddef>

// Problem constants (must match reference)
#define N0   1048576
#define N1_  65536
#define N2_  4096
#define DEG  16
#define IN_C 128
#define H1_  128
#define H2_  128
#define OUTC 64
#define C1_  256   // 2*H1
#define C2_  256   // 2*H2

typedef float v2f __attribute__((ext_vector_type(2)));
typedef float v8f __attribute__((ext_vector_type(8)));

static __device__ __forceinline__ v8f wmma_f32(v2f a, v2f b, v8f c) {
  // V_WMMA_F32_16X16X4_F32 : D = A(16x4) * B(4x16) + C(16x16), all f32
  return __builtin_amdgcn_wmma_f32_16x16x4_f32(
      /*neg_a=*/false, a, /*neg_b=*/false, b,
      /*c_mod=*/(short)0, c, /*reuse_a=*/false, /*reuse_b=*/false);
}

// Accumulate a 16x16 tile: A is a 16xK tile with contiguous rows (stride K, LDS),
// B points at weight row n0 (row-major OxK, stride K, global). r = lane&15,
// koff = (lane>>4)*2 selects the K-pair per the 16x4 f32 A layout.
template <int K>
static __device__ __forceinline__ v8f gemm_tile(const float* __restrict__ A,
                                                const float* __restrict__ B,
                                                int r, int koff, v8f acc) {
#pragma unroll
  for (int k0 = 0; k0 < K; k0 += 4) {
    v2f a, b;
    a.x = A[r * K + k0 + koff];
    a.y = A[r * K + k0 + koff + 1];
    b.x = B[r * K + k0 + koff];
    b.y = B[r * K + k0 + koff + 1];
    acc = wmma_f32(a, b, acc);
  }
  return acc;
}

// ---------------- Layer 0: fused gather + mean/max + 2xGEMM + bias + ReLU ----
__global__ __launch_bounds__(256) void sage_layer0(
    const float* __restrict__ x,   // N0 x 128
    const int*   __restrict__ ei0, // (2, N1*16), row0 = src
    const float* __restrict__ wlm, const float* __restrict__ blm,
    const float* __restrict__ wrm,
    const float* __restrict__ wlx, const float* __restrict__ blx,
    const float* __restrict__ wrx,
    float* __restrict__ h)         // N1 x 256
{
  __shared__ int   s_src[256];
  __shared__ float s_mean[16 * IN_C];
  __shared__ float s_max [16 * IN_C];
  __shared__ float s_xt  [16 * IN_C];

  const int tid  = threadIdx.x;
  const int row0 = blockIdx.x * 16;

  s_src[tid] = ei0[row0 * DEG + tid];   // 16 targets x 16 edges
  __syncthreads();

  const int wave = tid >> 5;
  const int lane = tid & 31;
  const int c4   = lane * 4;            // 4 channels per lane (128/32)

  for (int s = 0; s < 2; ++s) {
    const int tl = wave * 2 + s;        // local target 0..15
    float sx = 0.f, sy = 0.f, sz = 0.f, sw = 0.f;
    float mx = -3.402823466e38f, my = mx, mz = mx, mw = mx;
#pragma unroll
    for (int j = 0; j < DEG; ++j) {
      const int src = s_src[tl * DEG + j];
      const float4 v =
          *reinterpret_cast<const float4*>(x + (size_t)src * IN_C + c4);
      sx += v.x; sy += v.y; sz += v.z; sw += v.w;
      mx = fmaxf(mx, v.x); my = fmaxf(my, v.y);
      mz = fmaxf(mz, v.z); mw = fmaxf(mw, v.w);
    }
    float4 mean; mean.x = sx * 0.0625f; mean.y = sy * 0.0625f;
                 mean.z = sz * 0.0625f; mean.w = sw * 0.0625f;
    float4 mxv;  mxv.x = mx; mxv.y = my; mxv.z = mz; mxv.w = mw;
    *reinterpret_cast<float4*>(s_mean + tl * IN_C + c4) = mean;
    *reinterpret_cast<float4*>(s_max  + tl * IN_C + c4) = mxv;
    *reinterpret_cast<float4*>(s_xt   + tl * IN_C + c4) =
        *reinterpret_cast<const float4*>(x + (size_t)(row0 + tl) * IN_C + c4);
  }
  __syncthreads();

  const int r    = lane & 15;
  const int koff = (lane >> 4) * 2;
  const int n0   = wave * 16;           // output-col tile within a 128-wide half
  const int rowadd = (lane >> 4) * 8;

#pragma unroll
  for (int half = 0; half < 2; ++half) {
    const float* A1 = half ? s_max : s_mean;
    const float* B1 = (half ? wlx : wlm) + (size_t)n0 * IN_C;
    const float* B2 = (half ? wrx : wrm) + (size_t)n0 * IN_C;
    const float* bb = half ? blx : blm;
    v8f acc = {};
    acc = gemm_tile<IN_C>(A1,   B1, r, koff, acc);   // agg @ wl^T
    acc = gemm_tile<IN_C>(s_xt, B2, r, koff, acc);   // + x_tgt @ wr^T
    const float bv = bb[n0 + r];
#pragma unroll
    for (int rr = 0; rr < 8; ++rr) {
      const float val = fmaxf(acc[rr] + bv, 0.0f);   // ReLU(concat) fused
      h[(size_t)(row0 + rr + rowadd) * C1_ + half * H1_ + n0 + r] = val;
    }
  }
}

// ---------------- Layer 1: same, K = 256, no ReLU ---------------------------
__global__ __launch_bounds__(256) void sage_layer1(
    const float* __restrict__ hin, // N1 x 256
    const int*   __restrict__ ei1, // (2, N2*16)
    const float* __restrict__ wlm, const float* __restrict__ blm,
    const float* __restrict__ wrm,
    const float* __restrict__ wlx, const float* __restrict__ blx,
    const float* __restrict__ wrx,
    float* __restrict__ h2)        // N2 x 256
{
  __shared__ int   s_src[256];
  __shared__ float s_mean[16 * C1_];
  __shared__ float s_max [16 * C1_];
  __shared__ float s_ht  [16 * C1_];

  const int tid  = threadIdx.x;
  const int row0 = blockIdx.x * 16;

  s_src[tid] = ei1[row0 * DEG + tid];
  __syncthreads();

  const int wave = tid >> 5;
  const int lane = tid & 31;

  for (int s = 0; s < 2; ++s) {
    const int tl = wave * 2 + s;
#pragma unroll
    for (int p = 0; p < 2; ++p) {         // 256 channels = 2 chunks of 128
      const int c4 = p * 128 + lane * 4;
      float sx = 0.f, sy = 0.f, sz = 0.f, sw = 0.f;
      float mx = -3.402823466e38f, my = mx, mz = mx, mw = mx;
#pragma unroll
      for (int j = 0; j < DEG; ++j) {
        const int src = s_src[tl * DEG + j];
        const float4 v =
            *reinterpret_cast<const float4*>(hin + (size_t)src * C1_ + c4);
        sx += v.x; sy += v.y; sz += v.z; sw += v.w;
        mx = fmaxf(mx, v.x); my = fmaxf(my, v.y);
        mz = fmaxf(mz, v.z); mw = fmaxf(mw, v.w);
      }
      float4 mean; mean.x = sx * 0.0625f; mean.y = sy * 0.0625f;
                   mean.z = sz * 0.0625f; mean.w = sw * 0.0625f;
      float4 mxv;  mxv.x = mx; mxv.y = my; mxv.z = mz; mxv.w = mw;
      *reinterpret_cast<float4*>(s_mean + tl * C1_ + c4) = mean;
      *reinterpret_cast<float4*>(s_max  + tl * C1_ + c4) = mxv;
      *reinterpret_cast<float4*>(s_ht   + tl * C1_ + c4) =
          *reinterpret_cast<const float4*>(hin + (size_t)(row0 + tl) * C1_ + c4);
    }
  }
  __syncthreads();

  const int r    = lane & 15;
  const int koff = (lane >> 4) * 2;
  const int n0   = wave * 16;
  const int rowadd = (lane >> 4) * 8;

#pragma unroll
  for (int half = 0; half < 2; ++half) {
    const float* A1 = half ? s_max : s_mean;
    const float* B1 = (half ? wlx : wlm) + (size_t)n0 * C1_;
    const float* B2 = (half ? wrx : wrm) + (size_t)n0 * C1_;
    const float* bb = half ? blx : blm;
    v8f acc = {};
    acc = gemm_tile<C1_>(A1,   B1, r, koff, acc);
    acc = gemm_tile<C1_>(s_ht, B2, r, koff, acc);
    const float bv = bb[n0 + r];
#pragma unroll
    for (int rr = 0; rr < 8; ++rr) {
      h2[(size_t)(row0 + rr + rowadd) * C2_ + half * H2_ + n0 + r] =
          acc[rr] + bv;                    // no ReLU on layer 1
    }
  }
}

// ---------------- Post: logits = h2 @ w_post^T + b_post ---------------------
__global__ __launch_bounds__(128) void post_logits(
    const float* __restrict__ h2, const float* __restrict__ w,
    const float* __restrict__ b,  float* __restrict__ lg) // N2 x 64
{
  __shared__ float s_a[16 * C2_];
  const int tid  = threadIdx.x;
  const int row0 = blockIdx.x * 16;

  const float4* srcp = reinterpret_cast<const float4*>(h2 + (size_t)row0 * C2_);
  float4* dstp = reinterpret_cast<float4*>(s_a);
  for (int i = tid; i < 16 * C2_ / 4; i += 128) dstp[i] = srcp[i];
  __syncthreads();

  const int wave = tid >> 5;       // 4 waves -> 4 col-tiles of 16 (N=64)
  const int lane = tid & 31;
  const int r    = lane & 15;
  const int koff = (lane >> 4) * 2;
  const int n0   = wave * 16;
  const int rowadd = (lane >> 4) * 8;

  v8f acc = {};
  acc = gemm_tile<C2_>(s_a, w + (size_t)n0 * C2_, r, koff, acc);
  const float bv = b[n0 + r];
#pragma unroll
  for (int rr = 0; rr < 8; ++rr)
    lg[(size_t)(row0 + rr + rowadd) * OUTC + n0 + r] = acc[rr] + bv;
}

// ---------------- log_softmax over 64 cols, one wave per row ----------------
__global__ __launch_bounds__(256) void logsoftmax_k(
    const float* __restrict__ lg, float* __restrict__ out)
{
  const int lane = threadIdx.x & 31;
  const int row  = blockIdx.x * 8 + (threadIdx.x >> 5);
  const float* p = lg + (size_t)row * OUTC;
  const float v0 = p[lane];
  const float v1 = p[lane + 32];
  float m = fmaxf(v0, v1);
#pragma unroll
  for (int off = 16; off > 0; off >>= 1) m = fmaxf(m, __shfl_xor(m, off, 32));
  float ssum = expf(v0 - m) + expf(v1 - m);
#pragma unroll
  for (int off = 16; off > 0; off >>= 1) ssum += __shfl_xor(ssum, off, 32);
  const float lse = m + logf(ssum);
  out[(size_t)row * OUTC + lane]      = v0 - lse;
  out[(size_t)row * OUTC + lane + 32] = v1 - lse;
}

// ---------------- launcher ---------------------------------------------------
extern "C" void kernel_launch(void* const* d_in, const int* in_sizes, int n_in,
                              void* d_out, int out_size, void* d_ws,
                              size_t ws_size, hipStream_t stream) {
  (void)in_sizes; (void)n_in; (void)out_size; (void)ws_size;
  const float* x     = (const float*)d_in[0];
  const int*   ei0   = (const int*)d_in[1];
  const int*   ei1   = (const int*)d_in[2];
  const float* wl_m0 = (const float*)d_in[3];
  const float* bl_m0 = (const float*)d_in[4];
  const float* wr_m0 = (const float*)d_in[5];
  const float* wl_x0 = (const float*)d_in[6];
  const float* bl_x0 = (const float*)d_in[7];
  const float* wr_x0 = (const float*)d_in[8];
  const float* wl_m1 = (const float*)d_in[9];
  const float* bl_m1 = (const float*)d_in[10];
  const float* wr_m1 = (const float*)d_in[11];
  const float* wl_x1 = (const float*)d_in[12];
  const float* bl_x1 = (const float*)d_in[13];
  const float* wr_x1 = (const float*)d_in[14];
  const float* wpost = (const float*)d_in[15];
  const float* bpost = (const float*)d_in[16];

  float* ws = (float*)d_ws;
  float* h  = ws;                                   // N1*256 = 16,777,216 f
  float* h2 = ws + (size_t)N1_ * C1_;               // N2*256 =  1,048,576 f
  float* lg = h2 + (size_t)N2_ * C2_;               // N2*64  =    262,144 f

  sage_layer0<<<N1_ / 16, 256, 0, stream>>>(x, ei0, wl_m0, bl_m0, wr_m0,
                                            wl_x0, bl_x0, wr_x0, h);
  sage_layer1<<<N2_ / 16, 256, 0, stream>>>(h, ei1, wl_m1, bl_m1, wr_m1,
                                            wl_x1, bl_x1, wr_x1, h2);
  post_logits<<<N2_ / 16, 128, 0, stream>>>(h2, wpost, bpost, lg);
  logsoftmax_k<<<N2_ / 8, 256, 0, stream>>>(lg, (float*)d_out);
}